// CapsLayer2D_80118319940154
// MI455X (gfx1250) — compile-verified
//
#include <hip/hip_runtime.h>
#include <math.h>

typedef __attribute__((ext_vector_type(2))) float v2f;
typedef __attribute__((ext_vector_type(4))) float v4f;
typedef __attribute__((ext_vector_type(8))) float v8f;

#define P_TOTAL 6272   // 32*14*14 positions
#define NCAP 32        // n_input_caps  (== wave32 lanes)
#define ICAP 8         // input caps dim
#define KCAP 16        // n_caps
#define JCAP 16        // caps_dim
#define PT   16        // positions per block == WMMA M
#define LDSJ 20        // padded LDS row stride (floats) to break bank conflicts
#define SQ_EPS 1e-7f
#define NROUT 3

__global__ __launch_bounds__(256)
void caps_routing_kernel(const float* __restrict__ X,   // [P, NCAP, ICAP]
                         const float* __restrict__ W,   // [KCAP, NCAP, ICAP, JCAP]
                         float* __restrict__ Out)       // [P, KCAP, JCAP]
{
    __shared__ float pred[PT * NCAP * LDSJ];            // 40 KB

    const int tid   = threadIdx.x;
    const int lane  = tid & 31;
    const int wave  = tid >> 5;
    const int blk   = blockIdx.x;
    const int k     = blk & (KCAP - 1);
    const int ptile = blk >> 4;
    const int p0    = ptile * PT;

    // ---------- Stage A: predicted[p_local, n, j] for this k via f32 WMMA ----------
    // WMMA 16x16x4 f32: A is 16(pos) x 4(i), B is 4(i) x 16(j), D 16x16 f32.
    // Lane layout: m = lane&15 (row for A / col j for B,D), half = lane>>4 picks
    // i-pair {2h, 2h+1} per fragment register; D row = v + 8*half.
    const int m    = lane & 15;
    const int half = lane >> 4;

    for (int nn = 0; nn < 4; ++nn) {
        const int n = wave * 4 + nn;

        const float* xb = X + (size_t)(p0 + m) * (NCAP * ICAP) + n * ICAP;
        v2f a0, a1;                       // i0 = 0 and i0 = 4 fragments
        a0.x = xb[2 * half + 0];  a0.y = xb[2 * half + 1];
        a1.x = xb[2 * half + 4];  a1.y = xb[2 * half + 5];

        const float* wb = W + ((size_t)(k * NCAP + n) * ICAP) * JCAP + m;  // j = m
        v2f b0, b1;
        b0.x = wb[(2 * half + 0) * JCAP];  b0.y = wb[(2 * half + 1) * JCAP];
        b1.x = wb[(2 * half + 4) * JCAP];  b1.y = wb[(2 * half + 5) * JCAP];

        v8f acc = {};
        acc = __builtin_amdgcn_wmma_f32_16x16x4_f32(false, a0, false, b0,
                                                    (short)0, acc, false, false);
        acc = __builtin_amdgcn_wmma_f32_16x16x4_f32(false, a1, false, b1,
                                                    (short)0, acc, false, false);

        #pragma unroll
        for (int v = 0; v < 8; ++v) {
            const int prow = v + 8 * half;
            pred[(prow * NCAP + n) * LDSJ + m] = acc[v];
        }
    }
    __syncthreads();

    // ---------- Stage B: dynamic routing; lane == n (N=32 == wave32) ----------
    for (int pl = wave; pl < PT; pl += 8) {
        float pj[JCAP];
        #pragma unroll
        for (int j = 0; j < JCAP; ++j)
            pj[j] = pred[(pl * NCAP + lane) * LDSJ + j];

        float braw = 0.0f;
        float out[JCAP];

        for (int r = 0; r < NROUT; ++r) {
            // softmax over n (all 32 lanes)
            float mx = braw;
            #pragma unroll
            for (int s = 16; s >= 1; s >>= 1)
                mx = fmaxf(mx, __shfl_xor(mx, s, 32));
            float e = __expf(braw - mx);
            float se = e;
            #pragma unroll
            for (int s = 16; s >= 1; s >>= 1)
                se += __shfl_xor(se, s, 32);
            const float c = e / se;

            // out_j = sum_n c_n * pred_nj  (butterfly -> every lane has the sum)
            float sq = 0.0f;
            #pragma unroll
            for (int j = 0; j < JCAP; ++j) {
                float t = c * pj[j];
                #pragma unroll
                for (int s = 16; s >= 1; s >>= 1)
                    t += __shfl_xor(t, s, 32);
                out[j] = t;
                sq += t * t;
            }

            // squash: (sq/(1+sq)) * out / sqrt(sq + eps)
            const float scale = sq / ((1.0f + sq) * sqrtf(sq + SQ_EPS));

            // agreement update: b_n += sum_j pred_nj * out_j (per-lane)
            float dot = 0.0f;
            #pragma unroll
            for (int j = 0; j < JCAP; ++j) {
                out[j] *= scale;
                dot += pj[j] * out[j];
            }
            braw += dot;
        }

        if (lane == 0) {
            float* ob = Out + ((size_t)(p0 + pl) * KCAP + k) * JCAP;
            #pragma unroll
            for (int j = 0; j < JCAP; j += 4) {
                v4f t = { out[j], out[j + 1], out[j + 2], out[j + 3] };
                *(v4f*)(ob + j) = t;
            }
        }
    }
}

extern "C" void kernel_launch(void* const* d_in, const int* in_sizes, int n_in,
                              void* d_out, int out_size, void* d_ws, size_t ws_size,
                              hipStream_t stream) {
    const float* X = (const float*)d_in[0];   // [32,14,14,32,8] -> [6272,32,8]
    const float* W = (const float*)d_in[1];   // [16,32,8,16]
    float* Out = (float*)d_out;               // [6272,16,16]

    const int nblocks = (P_TOTAL / PT) * KCAP;   // 392 * 16 = 6272
    caps_routing_kernel<<<nblocks, 256, 0, stream>>>(X, W, Out);
}